// FocalLossAdaptive_32856499815078
// MI455X (gfx1250) — compile-verified
//
#include <hip/hip_runtime.h>
#include <hip/hip_bf16.h>
#include <float.h>
#include <stdint.h>

// Focal loss (adaptive gamma) on MI455X / gfx1250.
// Bandwidth-bound streaming reduction: async global->LDS double-buffered
// pipeline (ASYNCcnt) + online softmax in registers, per-row loss to d_ws,
// deterministic second-stage sum to d_out[0].

#define BLOCK 256
#define TILE  4096                      // floats per tile = 16 KB
#define CHUNKS (TILE / (BLOCK * 4))     // 4 x float4 chunks per thread per tile

#define GAMMA_LOW_THRESH 0.2f

__device__ __forceinline__ void async_copy_f4(uint32_t lds_addr, const float* g) {
  unsigned long long ga = (unsigned long long)(uintptr_t)g;
  // GLOBAL_LOAD_ASYNC_TO_LDS_B128: vdst = LDS byte address (VGPR),
  // addr = 64-bit global address (VGPR pair), saddr = off. Tracked by ASYNCcnt.
  asm volatile("global_load_async_to_lds_b128 %0, %1, off"
               :
               : "v"(lds_addr), "v"(ga)
               : "memory");
}

__device__ __forceinline__ void wait_async4() {
  asm volatile("s_wait_asynccnt 0x4" ::: "memory");
}
__device__ __forceinline__ void wait_async0() {
  asm volatile("s_wait_asynccnt 0x0" ::: "memory");
}

__global__ __launch_bounds__(BLOCK) void focal_row_kernel(
    const float* __restrict__ x, const int* __restrict__ target,
    float* __restrict__ row_loss, int C) {
  __shared__ __align__(16) float buf[2][TILE];
  __shared__ float rm[BLOCK / 32];
  __shared__ float rs[BLOCK / 32];

  const int tid = threadIdx.x;
  const int row = blockIdx.x;
  const float* rowp = x + (size_t)row * (size_t)C;

  // Target logit (only thread 0 needs it); issue early to hide latency.
  float xt = 0.0f;
  if (tid == 0) xt = rowp[target[row]];

  const int nt = (C + TILE - 1) / TILE;

  // Stage one tile: each thread async-copies its own CHUNKS float4s.
  // OOB chunks are clamped to the row base (always valid) and are simply
  // never consumed, keeping ASYNCcnt bookkeeping uniform (4 per stage).
  auto stage = [&](int t) {
    float* b = buf[t & 1];
#pragma unroll
    for (int j = 0; j < CHUNKS; ++j) {
      const int off = j * (BLOCK * 4) + tid * 4;
      const long idx = (long)t * TILE + off;
      const float* g = rowp + ((idx + 4 <= (long)C) ? idx : 0);
      const uint32_t l = (uint32_t)(uintptr_t)(&b[off]);
      async_copy_f4(l, g);
    }
  };

  // Online softmax accumulators (private per thread; no barriers needed
  // in the pipeline since each thread consumes exactly what it staged).
  float m = -FLT_MAX, s = 0.0f;

  stage(0);
  for (int t = 0; t < nt; ++t) {
    if (t + 1 < nt) {
      stage(t + 1);       // keep next tile in flight
      wait_async4();      // tile t (first 4 async ops) complete, in-order
    } else {
      wait_async0();
    }
    const float* b = buf[t & 1];
#pragma unroll
    for (int j = 0; j < CHUNKS; ++j) {
      const int off = j * (BLOCK * 4) + tid * 4;
      const long idx = (long)t * TILE + off;
      if (idx + 4 <= (long)C) {
        const float4 v = *(const float4*)(&b[off]);
        const float vmax = fmaxf(fmaxf(v.x, v.y), fmaxf(v.z, v.w));
        if (vmax > m) {               // rare after warmup -> ~1 exp/elem
          s *= __expf(m - vmax);
          m = vmax;
        }
        s += __expf(v.x - m) + __expf(v.y - m) +
             __expf(v.z - m) + __expf(v.w - m);
      }
    }
  }

  // Wave32 reduction of (m, s) pairs.
#pragma unroll
  for (int o = 16; o > 0; o >>= 1) {
    const float mo = __shfl_xor(m, o, 32);
    const float so = __shfl_xor(s, o, 32);
    const float mn = fmaxf(m, mo);
    s = s * __expf(m - mn) + so * __expf(mo - mn);
    m = mn;
  }
  if ((tid & 31) == 0) { rm[tid >> 5] = m; rs[tid >> 5] = s; }
  __syncthreads();

  if (tid == 0) {
    float M = rm[0], S = rs[0];
#pragma unroll
    for (int w = 1; w < BLOCK / 32; ++w) {
      const float mo = rm[w], so = rs[w];
      const float mn = fmaxf(M, mo);
      S = S * __expf(M - mn) + so * __expf(mo - mn);
      M = mn;
    }
    const float logpt = xt - M - logf(S);
    const float pt = expf(logpt);
    const float u = 1.0f - pt;
    const float u3 = u * u * u;          // gamma = 3 (pt >= 0.2)
    const float u5 = u3 * u * u;         // gamma = 5 (pt <  0.2)
    const float f = (pt < GAMMA_LOW_THRESH) ? u5 : u3;
    row_loss[row] = -f * logpt;
  }
}

// Deterministic single-block final reduction (no float atomics).
__global__ __launch_bounds__(1024) void reduce_sum_kernel(
    const float* __restrict__ v, float* __restrict__ out, int n) {
  __shared__ float sm[32];
  float acc = 0.0f;
  for (int i = threadIdx.x; i < n; i += 1024) acc += v[i];
#pragma unroll
  for (int o = 16; o > 0; o >>= 1) acc += __shfl_xor(acc, o, 32);
  if ((threadIdx.x & 31) == 0) sm[threadIdx.x >> 5] = acc;
  __syncthreads();
  if (threadIdx.x < 32) {
    float a = sm[threadIdx.x];           // 1024 threads -> exactly 32 waves
#pragma unroll
    for (int o = 16; o > 0; o >>= 1) a += __shfl_xor(a, o, 32);
    if (threadIdx.x == 0) out[0] = a;
  }
}

extern "C" void kernel_launch(void* const* d_in, const int* in_sizes, int n_in,
                              void* d_out, int out_size, void* d_ws, size_t ws_size,
                              hipStream_t stream) {
  const float* x = (const float*)d_in[0];     // [N, C] fp32 logits
  const int* target = (const int*)d_in[1];    // [N] int32 class indices
  float* out = (float*)d_out;                 // scalar
  float* ws = (float*)d_ws;                   // per-row losses (N floats)

  const int total = in_sizes[0];
  const int n = in_sizes[1];
  const int c = total / n;                    // 32000

  focal_row_kernel<<<n, BLOCK, 0, stream>>>(x, target, ws, c);
  reduce_sum_kernel<<<1, 1024, 0, stream>>>(ws, out, n);
}